// DeltaNet_31877247271542
// MI455X (gfx1250) — compile-verified
//
#include <hip/hip_runtime.h>

typedef float v2f __attribute__((ext_vector_type(2)));
typedef float v8f __attribute__((ext_vector_type(8)));
typedef float v4f __attribute__((ext_vector_type(4)));

#define CH   32     // chunk length
#define DK   128    // key dim
#define DVT  32     // dv tile per block
#define NCH  128    // chunks per sequence
#define LSEQ 4096
#define NBH  16     // b*h
#define NVT  4      // dv tiles (128/32)
#define LDK  132    // LDS stride for 32x128 tiles (pad vs 64 banks)
#define LD32 36     // LDS stride for 32-wide tiles / S slice

// ---------- gfx1250 async global->LDS path (guarded; falls back to sync copies) ----
#if defined(__gfx1250__) && __has_builtin(__builtin_amdgcn_global_load_async_to_lds_b128)
#define HAVE_ASYNC_LDS 1
#else
#define HAVE_ASYNC_LDS 0
#endif

// The builtin expects: (AS1 "int4*" global src, AS3 "int4*" LDS dst, imm offset, imm cpol)
typedef int b128_t __attribute__((vector_size(16)));
typedef __attribute__((address_space(1))) b128_t* glb_b128;
typedef __attribute__((address_space(3))) b128_t* lds_b128;

__device__ __forceinline__ void async_cp_b128(const float* g, float* l) {
#if HAVE_ASYNC_LDS
  glb_b128 gp = (glb_b128)(uintptr_t)g;                      // global flat == AS1
  lds_b128 lp = (lds_b128)(uintptr_t)(uint32_t)(uintptr_t)l; // low 32 bits = LDS offset
  __builtin_amdgcn_global_load_async_to_lds_b128(gp, lp, 0, 0);
#else
  *(v4f*)l = *(const v4f*)g;
#endif
}

__device__ __forceinline__ void wait_async0() {
#if HAVE_ASYNC_LDS
#if __has_builtin(__builtin_amdgcn_s_wait_asynccnt)
  __builtin_amdgcn_s_wait_asynccnt(0);
#else
  asm volatile("s_wait_asynccnt 0" ::: "memory");
#endif
#endif
}

// ---- wave-level 16x16 f32 WMMA tile: C += A(16xK) * B(Kx16), arbitrary strides ----
__device__ __forceinline__ v8f wmma_mm(const float* __restrict__ A, int ars, int acs,
                                       const float* __restrict__ B, int brs, int bcs,
                                       int K, v8f c) {
  const int lane = threadIdx.x & 31;
  const int mn = lane & 15;           // A row / B col for this lane
  const int ko = (lane >> 4) << 1;    // lanes 16-31 hold K+2,K+3
  #pragma unroll 4
  for (int k0 = 0; k0 < K; k0 += 4) {
    const int ka = k0 + ko;
    v2f a, b;
    a.x = A[mn * ars + ka * acs];
    a.y = A[mn * ars + (ka + 1) * acs];
    b.x = B[ka * brs + mn * bcs];
    b.y = B[(ka + 1) * brs + mn * bcs];
    c = __builtin_amdgcn_wmma_f32_16x16x4_f32(false, a, false, b, (short)0, c,
                                              false, false);
  }
  return c;
}

__device__ __forceinline__ v8f tile_load(const float* __restrict__ C, int ldc) {
  const int lane = threadIdx.x & 31;
  const int n = lane & 15;
  const int mb = (lane >> 4) << 3;
  v8f c;
  #pragma unroll
  for (int r = 0; r < 8; ++r) c[r] = C[(mb + r) * ldc + n];
  return c;
}

__device__ __forceinline__ void tile_store(float* __restrict__ C, int ldc, v8f d) {
  const int lane = threadIdx.x & 31;
  const int n = lane & 15;
  const int mb = (lane >> 4) << 3;
  #pragma unroll
  for (int r = 0; r < 8; ++r) C[(mb + r) * ldc + n] = d[r];
}

__global__ __launch_bounds__(256, 1)
void deltanet_scan_kernel(const float* __restrict__ qg, const float* __restrict__ kg,
                          const float* __restrict__ vg, const float* __restrict__ bg,
                          float* __restrict__ outg) {
  extern __shared__ float sm[];
  float* sQ    = sm;                    // 32 x LDK  (normalized q)
  float* sK    = sQ    + CH * LDK;      // 32 x LDK  (normalized k)
  float* sKB   = sK    + CH * LDK;      // 32 x LDK  (k_beta)
  float* sS    = sKB   + CH * LDK;      // 128 x LD32 (state slice)
  float* sV    = sS    + DK * LD32;     // 32 x LD32  (v*beta -> rhs)
  float* sP    = sV    + CH * LD32;     // 32 x LD32  (P -> A)
  float* sAttn = sP    + CH * LD32;     // 32 x LD32
  float* sR    = sAttn + CH * LD32;     // 32 x LD32  (running Tinv)
  float* sB    = sR    + CH * LD32;     // 32 x LD32  (A^(2^j))
  float* sT1   = sB    + CH * LD32;     // 32 x LD32  (temp / u_adj)
  float* sT2   = sT1   + CH * LD32;     // 32 x LD32  (temp / qS)
  float* sBeta = sT2   + CH * LD32;     // 32 (16B-aligned)
  float* sRq   = sBeta + CH;            // 32
  float* sRk   = sRq   + CH;            // 32
  float* sQss  = sRk   + CH;            // 32 x 8
  float* sKss  = sQss  + CH * 8;        // 32 x 8

  const int tid = threadIdx.x;
  const int wv  = tid >> 5;             // wave id 0..7 (wave32)
  const int bh  = blockIdx.y;           // 0..15
  const int v0  = blockIdx.x * DVT;     // dv column offset
  const size_t seq0 = (size_t)bh * LSEQ;

  for (int i = tid; i < DK * LD32; i += 256) sS[i] = 0.f;
  __syncthreads();

  #pragma unroll 1
  for (int c = 0; c < NCH; ++c) {
    const int l0 = c * CH;
    const int row = tid >> 3, seg = tid & 7;
    // ---------- async stage chunk into LDS (ASYNCcnt path) ----------
    {
      const size_t gro = (seq0 + l0 + row) * DK;
      const float* qp = qg + gro + seg * 16;
      const float* kp = kg + gro + seg * 16;
      float* qd = sQ + row * LDK + seg * 16;
      float* kd = sK + row * LDK + seg * 16;
      #pragma unroll
      for (int j = 0; j < 4; ++j) {
        async_cp_b128(qp + 4 * j, qd + 4 * j);
        async_cp_b128(kp + 4 * j, kd + 4 * j);
      }
      async_cp_b128(vg + gro + v0 + seg * 4, sV + row * LD32 + seg * 4);
      if (tid < 8) async_cp_b128(bg + seq0 + l0 + tid * 4, sBeta + tid * 4);
      if (c + 1 < NCH) {  // prefetch next chunk (global_prefetch_b8)
        const size_t nro = (seq0 + l0 + CH + row) * DK + seg * 16;
        __builtin_prefetch(qg + nro, 0, 3);
        __builtin_prefetch(kg + nro, 0, 3);
      }
      wait_async0();
    }
    __syncthreads();
    // ---------- partial row sum-squares (read back from LDS) ----------
    {
      const float* qd = sQ + row * LDK + seg * 16;
      const float* kd = sK + row * LDK + seg * 16;
      float aq = 0.f, ak = 0.f;
      #pragma unroll
      for (int i = 0; i < 16; ++i) {
        const float x = qd[i]; aq += x * x;
        const float y = kd[i]; ak += y * y;
      }
      sQss[row * 8 + seg] = aq;
      sKss[row * 8 + seg] = ak;
    }
    __syncthreads();
    // ---------- l2norm scales ----------
    if (tid < 64) {
      const int rr = tid & 31;
      const float* p = (tid < 32) ? sQss : sKss;
      float s = 0.f;
      #pragma unroll
      for (int j = 0; j < 8; ++j) s += p[rr * 8 + j];
      const float r = rsqrtf(s + 1e-6f);
      if (tid < 32) sRq[rr] = r; else sRk[rr] = r;
    }
    __syncthreads();
    // ---------- scale: qn, kn, k_beta, v*beta ----------
    {
      const float rq = sRq[row], rk = sRk[row], bt = sBeta[row];
      #pragma unroll
      for (int i = 0; i < 16; ++i) {
        const int idx = row * LDK + seg * 16 + i;
        sQ[idx] *= rq;
        const float kk = sK[idx] * rk;
        sK[idx] = kk;
        sKB[idx] = kk * bt;
      }
      #pragma unroll
      for (int i = 0; i < 4; ++i) sV[row * LD32 + seg * 4 + i] *= bt;
    }
    __syncthreads();
    // ---------- phase: P = kb*kn^T (w0-3) || attn = qn*kn^T (w4-7), K=128 ----------
    {
      const int t = wv & 3, ti = t >> 1, tj = t & 1;
      v8f acc = {};
      if (wv < 4) {
        acc = wmma_mm(sKB + ti * 16 * LDK, LDK, 1, sK + tj * 16 * LDK, 1, LDK, DK, acc);
        tile_store(sP + ti * 16 * LD32 + tj * 16, LD32, acc);
      } else {
        acc = wmma_mm(sQ + ti * 16 * LDK, LDK, 1, sK + tj * 16 * LDK, 1, LDK, DK, acc);
        tile_store(sAttn + ti * 16 * LD32 + tj * 16, LD32, acc);
      }
    }
    __syncthreads();
    // ---------- masks: A = strict_lower(-P); R = I + A; attn lower-incl ----------
    for (int i = tid; i < CH * CH; i += 256) {
      const int r = i >> 5, cc = i & 31, idx = r * LD32 + cc;
      const float a = (r > cc) ? -sP[idx] : 0.f;
      sP[idx] = a;
      sR[idx] = a + ((r == cc) ? 1.f : 0.f);
      sAttn[idx] = (cc > r) ? 0.f : sAttn[idx];
    }
    __syncthreads();
    // ---------- phase: B = A*A (w0-3) || T1 = kb @ S, K=128 (w4-7) ----------
    {
      const int t = wv & 3, ti = t >> 1, tj = t & 1;
      v8f acc = {};
      if (wv < 4) {
        acc = wmma_mm(sP + ti * 16 * LD32, LD32, 1, sP + tj * 16, LD32, 1, CH, acc);
        tile_store(sB + ti * 16 * LD32 + tj * 16, LD32, acc);
      } else {
        acc = wmma_mm(sKB + ti * 16 * LDK, LDK, 1, sS + tj * 16, LD32, 1, DK, acc);
        tile_store(sT1 + ti * 16 * LD32 + tj * 16, LD32, acc);
      }
    }
    __syncthreads();
    // ---------- rhs = v*beta - kb@S ----------
    for (int i = tid; i < CH * CH; i += 256) {
      const int r = i >> 5, cc = i & 31;
      sV[r * LD32 + cc] -= sT1[r * LD32 + cc];
    }
    __syncthreads();
    // ---------- Neumann doubling: R *= (I + A^(2^j)); B squares alongside ----------
    #pragma unroll 1
    for (int it = 0; it < 3; ++it) {
      const int t = wv & 3, ti = t >> 1, tj = t & 1;
      v8f acc = {};
      if (wv < 4) {
        acc = wmma_mm(sR + ti * 16 * LD32, LD32, 1, sB + tj * 16, LD32, 1, CH, acc);
        tile_store(sT1 + ti * 16 * LD32 + tj * 16, LD32, acc);
      } else {
        acc = wmma_mm(sB + ti * 16 * LD32, LD32, 1, sB + tj * 16, LD32, 1, CH, acc);
        tile_store(sT2 + ti * 16 * LD32 + tj * 16, LD32, acc);
      }
      __syncthreads();
      for (int i = tid; i < CH * CH; i += 256) {
        const int r = i >> 5, cc = i & 31, idx = r * LD32 + cc;
        sR[idx] += sT1[idx];
        sB[idx] = sT2[idx];
      }
      __syncthreads();
    }
    // ---------- final factor (w0-3) || T2 = qn @ S, K=128 (w4-7) ----------
    {
      const int t = wv & 3, ti = t >> 1, tj = t & 1;
      v8f acc = {};
      if (wv < 4) {
        acc = wmma_mm(sR + ti * 16 * LD32, LD32, 1, sB + tj * 16, LD32, 1, CH, acc);
        tile_store(sT1 + ti * 16 * LD32 + tj * 16, LD32, acc);
      } else {
        acc = wmma_mm(sQ + ti * 16 * LDK, LDK, 1, sS + tj * 16, LD32, 1, DK, acc);
        tile_store(sT2 + ti * 16 * LD32 + tj * 16, LD32, acc);
      }
    }
    __syncthreads();
    for (int i = tid; i < CH * CH; i += 256) {
      const int r = i >> 5, cc = i & 31;
      sR[r * LD32 + cc] += sT1[r * LD32 + cc];   // sR = Tinv
    }
    __syncthreads();
    // ---------- u_adj = Tinv @ rhs -> sT1 ----------
    if (wv < 4) {
      const int t = wv & 3, ti = t >> 1, tj = t & 1;
      v8f acc = {};
      acc = wmma_mm(sR + ti * 16 * LD32, LD32, 1, sV + tj * 16, LD32, 1, CH, acc);
      tile_store(sT1 + ti * 16 * LD32 + tj * 16, LD32, acc);
    }
    __syncthreads();
    // ---------- o = qS + attn@u_adj -> global (w0-3) || S tiles 0-7 (w4-7) ----------
    if (wv < 4) {
      const int ti = wv >> 1, tj = wv & 1;
      v8f acc = tile_load(sT2 + ti * 16 * LD32 + tj * 16, LD32);
      acc = wmma_mm(sAttn + ti * 16 * LD32, LD32, 1, sT1 + tj * 16, LD32, 1, CH, acc);
      const int lane = tid & 31, n = lane & 15, mb = (lane >> 4) << 3;
      const size_t base = (seq0 + l0 + ti * 16 + mb) * DK + v0 + tj * 16 + n;
      #pragma unroll
      for (int r8 = 0; r8 < 8; ++r8) outg[base + (size_t)r8 * DK] = acc[r8];
    } else {
      #pragma unroll
      for (int s = 0; s < 2; ++s) {
        const int t2 = (wv - 4) * 2 + s;            // S = S + kn^T @ u_adj
        const int mi = t2 >> 1, nj = t2 & 1;
        v8f acc = tile_load(sS + mi * 16 * LD32 + nj * 16, LD32);
        acc = wmma_mm(sK + mi * 16, 1, LDK, sT1 + nj * 16, LD32, 1, CH, acc);
        tile_store(sS + mi * 16 * LD32 + nj * 16, LD32, acc);
      }
    }
    __syncthreads();
    // ---------- S tiles 8-15, one per wave ----------
    {
      const int t2 = 8 + wv;
      const int mi = t2 >> 1, nj = t2 & 1;
      v8f acc = tile_load(sS + mi * 16 * LD32 + nj * 16, LD32);
      acc = wmma_mm(sK + mi * 16, 1, LDK, sT1 + nj * 16, LD32, 1, CH, acc);
      tile_store(sS + mi * 16 * LD32 + nj * 16, LD32, acc);
    }
    __syncthreads();
  }
  // ---------- write final state S ----------
  const size_t sbase = (size_t)NBH * LSEQ * DK + (size_t)bh * DK * DK + v0;
  for (int i = tid; i < DK * DVT; i += 256) {
    const int r = i >> 5, cc = i & 31;
    outg[sbase + (size_t)r * DK + cc] = sS[r * LD32 + cc];
  }
}

extern "C" void kernel_launch(void* const* d_in, const int* in_sizes, int n_in,
                              void* d_out, int out_size, void* d_ws, size_t ws_size,
                              hipStream_t stream) {
  (void)in_sizes; (void)n_in; (void)out_size; (void)d_ws; (void)ws_size;
  const float* q    = (const float*)d_in[0];
  const float* k    = (const float*)d_in[1];
  const float* v    = (const float*)d_in[2];
  const float* beta = (const float*)d_in[3];
  float* out = (float*)d_out;

  const size_t smem_floats =
      3 * (size_t)CH * LDK +        // sQ, sK, sKB
      (size_t)DK * LD32 +           // sS
      7 * (size_t)CH * LD32 +       // sV, sP, sAttn, sR, sB, sT1, sT2
      3 * CH +                      // sBeta, sRq, sRk
      2 * CH * 8;                   // sQss, sKss
  const size_t smem_bytes = smem_floats * sizeof(float);  // ~101 KB (LDS: 320 KB/WGP)

  (void)hipFuncSetAttribute((const void*)deltanet_scan_kernel,
                            hipFuncAttributeMaxDynamicSharedMemorySize,
                            (int)smem_bytes);

  dim3 grid(NVT, NBH);   // 4 dv-tiles x 16 (b*h) = 64 blocks
  dim3 block(256);       // 8 wave32 waves
  deltanet_scan_kernel<<<grid, block, smem_bytes, stream>>>(q, k, v, beta, out);
}